// IPA_Module_8538394985044
// MI455X (gfx1250) — compile-verified
//
#include <hip/hip_runtime.h>
#include <math.h>

// ---- problem constants (from reference) ----
#define B_  2
#define R_  384
#define CM_ 384
#define CZ_ 128
#define H_  12
#define DH_ 32
#define NQP_ 4

#define W_L_ 1.7320508075688772f          /* sqrt(3) */
#define W_C_ 4.242640687119285f           /* sqrt(18) = (2/(9*4))^-0.5 */
#define INV_SQRT_DH_ 0.17677669529663687f /* 32^-0.5 */

// LDS row strides padded so (stride mod 64) != 0  -> conflict-free ds loads
#define LROW_  388   /* for 384-wide tiles */
#define LROWB_ 132   /* for 128-wide tiles */

typedef __attribute__((ext_vector_type(2))) float v2f;
typedef __attribute__((ext_vector_type(8))) float v8f;
typedef __attribute__((ext_vector_type(4))) int   v4i;

#define GLOBAL_AS __attribute__((address_space(1)))
#define LDS_AS    __attribute__((address_space(3)))

// D = A(16x4) * B(4x16) + C  (f32 WMMA, wave32)
static __device__ inline v8f wmma4(v2f a, v2f b, v8f c) {
    return __builtin_amdgcn_wmma_f32_16x16x4_f32(
        /*neg_a=*/false, a, /*neg_b=*/false, b,
        /*c_mod=*/(short)0, c, /*reuse_a=*/false, /*reuse_b=*/false);
}

// ---- CDNA5 async global->LDS copy (guarded; falls back to ld/st) ----
#if defined(__has_builtin)
#if __has_builtin(__builtin_amdgcn_global_load_async_to_lds_b128) && \
    __has_builtin(__builtin_amdgcn_s_wait_asynccnt)
#define HAVE_ASYNC_LDS 1
#endif
#endif

static __device__ inline void cp16_g2l(const float* g, float* l) {
#ifdef HAVE_ASYNC_LDS
    __builtin_amdgcn_global_load_async_to_lds_b128(
        (GLOBAL_AS v4i*)(GLOBAL_AS const void*)g,
        (LDS_AS v4i*)(LDS_AS void*)l, 0, 0);
#else
    float4 v = *(const float4*)g;
    *(float4*)l = v;
#endif
}

static __device__ inline void cp_wait_all() {
#ifdef HAVE_ASYNC_LDS
    __builtin_amdgcn_s_wait_asynccnt(0);
#endif
    __syncthreads();
}

// ---------------------------------------------------------------------------
// Generic f32 GEMM: C[M,N] = A[M,K] @ B[K,N].  One wave per 16x16 C tile.
// ---------------------------------------------------------------------------
__global__ __launch_bounds__(32) void gemm_f32_wmma(
    const float* __restrict__ A, const float* __restrict__ Bm,
    float* __restrict__ C, int M, int N, int K) {
    int ntn  = (N + 15) >> 4;
    int tile = blockIdx.x;
    int tm = tile / ntn, tn = tile % ntn;
    int lane = threadIdx.x & 31;
    int hi = lane >> 4, lo = lane & 15;

    int row  = tm * 16 + lo;
    int col  = tn * 16 + lo;
    int colc = (col < N) ? col : (N - 1);
    bool cok = (col < N);

    v8f c = {};
    for (int k0 = 0; k0 < K; k0 += 4) {
        int ka = k0 + 2 * hi;
        v2f a, b;
        a.x = A[(size_t)row * K + ka];
        a.y = A[(size_t)row * K + ka + 1];
        float bx = Bm[(size_t)ka * N + colc];
        float by = Bm[(size_t)(ka + 1) * N + colc];
        b.x = cok ? bx : 0.0f;
        b.y = cok ? by : 0.0f;
        c = wmma4(a, b, c);
    }
    for (int r = 0; r < 8; ++r) {
        int orow = tm * 16 + r + hi * 8;
        if (cok) C[(size_t)orow * N + col] = c[r];
    }
}

// ---------------------------------------------------------------------------
// Rearrange qkv/qk, apply backbone frames, compute point norms.
// ---------------------------------------------------------------------------
__global__ void prep_kernel(
    const float* __restrict__ qkv_raw, const float* __restrict__ qk_raw,
    const float* __restrict__ bbr, const float* __restrict__ bbt,
    float* __restrict__ rq, float* __restrict__ rk, float* __restrict__ rv,
    float* __restrict__ Tq, float* __restrict__ Tk,
    float* __restrict__ qn, float* __restrict__ kn) {
    int idx = blockIdx.x * blockDim.x + threadIdx.x;   // over B*H*R
    if (idx >= B_ * H_ * R_) return;
    int r  = idx % R_;
    int bh = idx / R_;
    int h  = bh % H_;
    int b  = bh / H_;

    const float* qrow = qkv_raw + (size_t)(b * R_ + r) * (3 * H_ * DH_);
    float* rqp = rq + (size_t)idx * DH_;
    float* rkp = rk + (size_t)idx * DH_;
    float* rvp = rv + (size_t)idx * DH_;
    for (int d = 0; d < DH_; ++d) {
        rqp[d] = qrow[d * 3 * H_ + h] * INV_SQRT_DH_;   // fold dot scale into q
        rkp[d] = qrow[d * 3 * H_ + H_ + h];
        rvp[d] = qrow[d * 3 * H_ + 2 * H_ + h];
    }

    const float* krow = qk_raw + (size_t)(b * R_ + r) * (H_ * 2 * NQP_ * 3);
    const float* Rm   = bbr + (size_t)(b * R_ + r) * 9;   // [a][k]
    const float* tv   = bbt + (size_t)(b * R_ + r) * 3;
    float accq = 0.0f, acck = 0.0f;
    for (int p = 0; p < NQP_; ++p) {
        const float* gq = krow + ((h * 2 + 0) * NQP_ + p) * 3;
        const float* gk = krow + ((h * 2 + 1) * NQP_ + p) * 3;
        for (int k = 0; k < 3; ++k) {
            float sq = tv[k], sk = tv[k];
            for (int a = 0; a < 3; ++a) {
                sq += gq[a] * Rm[a * 3 + k];
                sk += gk[a] * Rm[a * 3 + k];
            }
            Tq[(size_t)idx * 12 + p * 3 + k] = sq;  accq += sq * sq;
            Tk[(size_t)idx * 12 + p * 3 + k] = sk;  acck += sk * sk;
        }
    }
    qn[idx] = accq;
    kn[idx] = acck;
}

// ---------------------------------------------------------------------------
// pair_bias: (B*R*R, CZ) @ (CZ, H) -> scattered to (B,H,R,R). N=12 pad 16.
// One wave per 16-row M tile; A tile staged to LDS via async b128 copies.
// ---------------------------------------------------------------------------
__global__ __launch_bounds__(32) void bias_kernel(
    const float* __restrict__ pair, const float* __restrict__ Wfc1,
    const float* __restrict__ bfc1, float* __restrict__ biasb) {
    __shared__ float At[16 * LROWB_];     // 16x128 A tile, padded rows

    int tm   = blockIdx.x;                // 0 .. (B*R*R/16 - 1)
    int lane = threadIdx.x & 31;
    int hi = lane >> 4, lo = lane & 15;

    const float* Ab = pair + (size_t)tm * 16 * CZ_;
    // stage: row r, lane loads float4 #lane (128 floats = 32 float4 per row)
    for (int r = 0; r < 16; ++r)
        cp16_g2l(Ab + (size_t)r * CZ_ + lane * 4, At + r * LROWB_ + lane * 4);
    cp_wait_all();

    bool cok = (lo < H_);
    int colc = cok ? lo : (H_ - 1);

    v8f c = {};
    for (int k0 = 0; k0 < CZ_; k0 += 4) {
        int ka = k0 + 2 * hi;
        v2f a, b;
        a.x = At[lo * LROWB_ + ka];
        a.y = At[lo * LROWB_ + ka + 1];
        float bx = Wfc1[ka * H_ + colc];
        float by = Wfc1[(ka + 1) * H_ + colc];
        b.x = cok ? bx : 0.0f;
        b.y = cok ? by : 0.0f;
        c = wmma4(a, b, c);
    }

    // tile of 16 consecutive flat (b,i,j) rows lies inside one (b,i): j fastest
    size_t base = (size_t)tm * 16;
    int bb = (int)(base / ((size_t)R_ * R_));
    int rem = (int)(base % ((size_t)R_ * R_));
    int i = rem / R_, j0 = rem % R_;
    if (cok) {
        float bias_h = bfc1[lo];
        float* dst = biasb + (((size_t)(bb * H_ + lo) * R_ + i) * R_) + j0;
        for (int r = 0; r < 8; ++r)
            dst[r + hi * 8] = c[r] + bias_h;
    }
}

// ---------------------------------------------------------------------------
// Fused attention: one wave per (b,h,i-tile).
// ---------------------------------------------------------------------------
__global__ __launch_bounds__(32) void attn_kernel(
    const float* __restrict__ rq, const float* __restrict__ rk,
    const float* __restrict__ rv,
    const float* __restrict__ Tq, const float* __restrict__ Tk,
    const float* __restrict__ qn, const float* __restrict__ kn,
    const float* __restrict__ biasb, const float* __restrict__ gamma,
    float* __restrict__ attn_out, float* __restrict__ avout) {
    __shared__ float Lg[16 * LROW_];     // logits/attn tile (padded rows)

    int blk   = blockIdx.x;              // (b*H + h) * (R/16) + itile
    int itile = blk % (R_ / 16);
    int bh    = blk / (R_ / 16);
    int h     = bh % H_;
    int b     = bh / H_;
    int lane  = threadIdx.x & 31;
    int hi = lane >> 4, lo = lane & 15;
    int i0 = itile * 16;

    float head_w = logf(1.0f + expf(gamma[0])) * (W_C_ * 0.5f);

    const float* rqb = rq + (size_t)bh * R_ * DH_;
    const float* rkb = rk + (size_t)bh * R_ * DH_;
    const float* rvb = rv + (size_t)bh * R_ * DH_;
    const float* Tqb = Tq + (size_t)bh * R_ * 12;
    const float* Tkb = Tk + (size_t)bh * R_ * 12;
    const float* qnb = qn + (size_t)bh * R_;
    const float* knb = kn + (size_t)bh * R_;
    const float* bsb = biasb + (size_t)bh * R_ * R_ + (size_t)i0 * R_;

    for (int jt = 0; jt < R_ / 16; ++jt) {
        int j0 = jt * 16;
        v8f cdot = {}, ccr = {};
        for (int k0 = 0; k0 < DH_; k0 += 4) {            // scalar dot, K=32
            int ka = k0 + 2 * hi;
            v2f a, bm;
            a.x  = rqb[(size_t)(i0 + lo) * DH_ + ka];
            a.y  = rqb[(size_t)(i0 + lo) * DH_ + ka + 1];
            bm.x = rkb[(size_t)(j0 + lo) * DH_ + ka];    // B = rk^T
            bm.y = rkb[(size_t)(j0 + lo) * DH_ + ka + 1];
            cdot = wmma4(a, bm, cdot);
        }
        for (int k0 = 0; k0 < 12; k0 += 4) {             // point cross, K=12
            int ka = k0 + 2 * hi;
            v2f a, bm;
            a.x  = Tqb[(size_t)(i0 + lo) * 12 + ka];
            a.y  = Tqb[(size_t)(i0 + lo) * 12 + ka + 1];
            bm.x = Tkb[(size_t)(j0 + lo) * 12 + ka];
            bm.y = Tkb[(size_t)(j0 + lo) * 12 + ka + 1];
            ccr = wmma4(a, bm, ccr);
        }
        for (int r = 0; r < 8; ++r) {
            int i = r + hi * 8;
            int j = j0 + lo;
            float sq = qnb[i0 + i] + knb[j] - 2.0f * ccr[r];
            Lg[i * LROW_ + j] = W_L_ * (bsb[(size_t)i * R_ + j] + cdot[r] + head_w * sq);
        }
    }
    __syncthreads();

    // softmax: 2 lanes per row (lane lo owns row lo; hi selects half)
    {
        int jb = hi * (R_ / 2), je = jb + (R_ / 2);
        float m = -1e30f;
        for (int j = jb; j < je; ++j) m = fmaxf(m, Lg[lo * LROW_ + j]);
        m = fmaxf(m, __shfl_xor(m, 16));
        float s = 0.0f;
        for (int j = jb; j < je; ++j) {
            float e = __expf(Lg[lo * LROW_ + j] - m);
            Lg[lo * LROW_ + j] = e;
            s += e;
        }
        s += __shfl_xor(s, 16);
        float inv = 1.0f / s;
        for (int j = jb; j < je; ++j) Lg[lo * LROW_ + j] *= inv;
    }
    __syncthreads();

    // spill attn tile for the pair-aggregation kernel
    float* aout = attn_out + (size_t)bh * R_ * R_ + (size_t)i0 * R_;
    for (int i = 0; i < 16; ++i)
        for (int j = lane; j < R_; j += 32)
            aout[(size_t)i * R_ + j] = Lg[i * LROW_ + j];

    // attn @ v : (16 x 384) @ (384 x 32), two N tiles
    for (int nt = 0; nt < 2; ++nt) {
        v8f c = {};
        for (int k0 = 0; k0 < R_; k0 += 4) {
            int ka = k0 + 2 * hi;
            v2f a, bm;
            a.x  = Lg[lo * LROW_ + ka];
            a.y  = Lg[lo * LROW_ + ka + 1];
            bm.x = rvb[(size_t)ka * DH_ + nt * 16 + lo];
            bm.y = rvb[(size_t)(ka + 1) * DH_ + nt * 16 + lo];
            c = wmma4(a, bm, c);
        }
        for (int r = 0; r < 8; ++r) {
            int i = i0 + r + hi * 8;
            int d = nt * 16 + lo;
            avout[((size_t)(b * R_ + i)) * (H_ * DH_) + h * DH_ + d] = c[r];
        }
    }
}

// ---------------------------------------------------------------------------
// top: per (b,i): attn rows (Hpad16 x R) @ pair_rep[b,i] (R x CZ).
// One 256-thread block per (b,i); attn rows staged in LDS; 8 waves = 8 n-tiles.
// ---------------------------------------------------------------------------
__global__ __launch_bounds__(256) void top_kernel(
    const float* __restrict__ attn, const float* __restrict__ pair,
    float* __restrict__ topPre) {
    __shared__ float Aat[16 * LROW_];    // attn rows h=0..11, pad 12..15 zero

    int bi  = blockIdx.x;
    int b   = bi / R_, i = bi % R_;
    int tid = threadIdx.x;

    // stage H_*R_ floats (h rows contiguous in global) as float4 copies
    for (int t = tid; t < (H_ * R_) / 4; t += 256) {
        int h = t / (R_ / 4);
        int j = (t % (R_ / 4)) * 4;
        cp16_g2l(attn + (((size_t)(b * H_ + h) * R_ + i) * R_) + j,
                 Aat + h * LROW_ + j);
    }
    for (int t = tid; t < (4 * R_) / 4; t += 256) {      // zero pad rows 12..15
        int h = 12 + t / (R_ / 4);
        int j = (t % (R_ / 4)) * 4;
        float4 z = {0.0f, 0.0f, 0.0f, 0.0f};
        *(float4*)(Aat + h * LROW_ + j) = z;
    }
    cp_wait_all();

    int wave = tid >> 5;                 // n-tile 0..7
    int lane = tid & 31;
    int hi = lane >> 4, lo = lane & 15;
    const float* pbase = pair + (size_t)bi * R_ * CZ_;

    // prefetch this (b,i)'s pair slice (second pass -> L2 resident anyway)
    if (wave == 0)
        for (int kp = lane; kp < R_; kp += 32)
            __builtin_prefetch(pbase + (size_t)kp * CZ_, 0, 3);

    v8f c = {};
    for (int k0 = 0; k0 < R_; k0 += 4) {
        int ka = k0 + 2 * hi;
        v2f a, bm;
        a.x  = Aat[lo * LROW_ + ka];
        a.y  = Aat[lo * LROW_ + ka + 1];
        bm.x = pbase[(size_t)ka * CZ_ + wave * 16 + lo];
        bm.y = pbase[(size_t)(ka + 1) * CZ_ + wave * 16 + lo];
        c = wmma4(a, bm, c);
    }
    for (int r = 0; r < 8; ++r) {
        int h = r + hi * 8;
        if (h < H_)
            topPre[(size_t)bi * (H_ * CZ_) + h * CZ_ + wave * 16 + lo] = c[r];
    }
}

// ---------------------------------------------------------------------------
// out = av(768x384) @ W0(384x384) + topPre(768x1536) @ W1(1536x384)
// ---------------------------------------------------------------------------
__global__ __launch_bounds__(32) void out_kernel(
    const float* __restrict__ av, const float* __restrict__ W0,
    const float* __restrict__ topPre, const float* __restrict__ W1,
    float* __restrict__ out) {
    int blk = blockIdx.x;
    int tn  = blk % (CM_ / 16);
    int tm  = blk / (CM_ / 16);
    int lane = threadIdx.x & 31;
    int hi = lane >> 4, lo = lane & 15;
    int row = tm * 16 + lo, col = tn * 16 + lo;

    v8f c = {};
    for (int k0 = 0; k0 < H_ * DH_; k0 += 4) {           // K = 384
        int ka = k0 + 2 * hi;
        v2f a, bm;
        a.x  = av[(size_t)row * (H_ * DH_) + ka];
        a.y  = av[(size_t)row * (H_ * DH_) + ka + 1];
        bm.x = W0[(size_t)ka * CM_ + col];
        bm.y = W0[(size_t)(ka + 1) * CM_ + col];
        c = wmma4(a, bm, c);
    }
    for (int k0 = 0; k0 < H_ * CZ_; k0 += 4) {           // K = 1536
        int ka = k0 + 2 * hi;
        v2f a, bm;
        a.x  = topPre[(size_t)row * (H_ * CZ_) + ka];
        a.y  = topPre[(size_t)row * (H_ * CZ_) + ka + 1];
        bm.x = W1[(size_t)ka * CM_ + col];
        bm.y = W1[(size_t)(ka + 1) * CM_ + col];
        c = wmma4(a, bm, c);
    }
    for (int r = 0; r < 8; ++r)
        out[(size_t)(tm * 16 + r + hi * 8) * CM_ + tn * 16 + lo] = c[r];
}

// ---------------------------------------------------------------------------
extern "C" void kernel_launch(void* const* d_in, const int* in_sizes, int n_in,
                              void* d_out, int out_size, void* d_ws, size_t ws_size,
                              hipStream_t stream) {
    const float* pair  = (const float*)d_in[0];
    const float* sing  = (const float*)d_in[1];
    const float* bbr   = (const float*)d_in[2];
    const float* bbt   = (const float*)d_in[3];
    const float* Wqkv  = (const float*)d_in[4];
    const float* Wqk   = (const float*)d_in[5];
    const float* Wfc1  = (const float*)d_in[6];
    const float* bfc1  = (const float*)d_in[7];
    const float* W0    = (const float*)d_in[8];
    const float* W1    = (const float*)d_in[9];
    const float* gamma = (const float*)d_in[10];
    float* out = (float*)d_out;

    float* ws = (float*)d_ws;
    size_t off = 0;
    float* qkv_raw = ws + off; off += (size_t)B_ * R_ * (3 * H_ * DH_);
    float* qk_raw  = ws + off; off += (size_t)B_ * R_ * (H_ * 2 * NQP_ * 3);
    float* rq      = ws + off; off += (size_t)B_ * H_ * R_ * DH_;
    float* rk      = ws + off; off += (size_t)B_ * H_ * R_ * DH_;
    float* rv      = ws + off; off += (size_t)B_ * H_ * R_ * DH_;
    float* Tqb     = ws + off; off += (size_t)B_ * H_ * R_ * 12;
    float* Tkb     = ws + off; off += (size_t)B_ * H_ * R_ * 12;
    float* qnb     = ws + off; off += (size_t)B_ * H_ * R_;
    float* knb     = ws + off; off += (size_t)B_ * H_ * R_;
    float* biasb   = ws + off; off += (size_t)B_ * H_ * R_ * R_;
    float* attnb   = ws + off; off += (size_t)B_ * H_ * R_ * R_;
    float* avb     = ws + off; off += (size_t)B_ * R_ * H_ * DH_;
    float* topPre  = ws + off; off += (size_t)B_ * R_ * H_ * CZ_;

    const int MR = B_ * R_;                                  // 768 rows
    gemm_f32_wmma<<<(MR / 16) * ((3 * H_ * DH_) / 16), 32, 0, stream>>>(
        sing, Wqkv, qkv_raw, MR, 3 * H_ * DH_, CM_);
    gemm_f32_wmma<<<(MR / 16) * ((H_ * 2 * NQP_ * 3) / 16), 32, 0, stream>>>(
        sing, Wqk, qk_raw, MR, H_ * 2 * NQP_ * 3, CM_);
    prep_kernel<<<(B_ * H_ * R_ + 255) / 256, 256, 0, stream>>>(
        qkv_raw, qk_raw, bbr, bbt, rq, rk, rv, Tqb, Tkb, qnb, knb);
    bias_kernel<<<(B_ * R_ * R_) / 16, 32, 0, stream>>>(pair, Wfc1, bfc1, biasb);
    attn_kernel<<<B_ * H_ * (R_ / 16), 32, 0, stream>>>(
        rq, rk, rv, Tqb, Tkb, qnb, knb, biasb, gamma, attnb, avb);
    top_kernel<<<B_ * R_, 256, 0, stream>>>(attnb, pair, topPre);
    out_kernel<<<(MR / 16) * (CM_ / 16), 32, 0, stream>>>(avb, W0, topPre, W1, out);
}